// Attention_50740743635025
// MI455X (gfx1250) — compile-verified
//
#include <hip/hip_runtime.h>
#include <cmath>

// ---------------------------------------------------------------------------
// MI455X (gfx1250, wave32) implementation.
// Problem: C=10, B=8192, F=256, O=64, DK=5, all f32.
// Memory-bound (~30 MB HBM traffic vs 2.68 GFLOP) -> keep exact f32 and use
// V_WMMA_F32_16X16X4_F32 for the dominant [8192x256]x[256x64] GEMMs.
//
// Round-1 refinement: LDS weight tile uses a paired-K layout so each WMMA
// B-fragment (rows k,k+1 at one column) is a contiguous 8B-aligned pair in
// LDS -> single ds_load_b64 straight into the WMMA operand register pair,
// removing the v_dual_mov shuffle seen in the round-0 disassembly.
// ---------------------------------------------------------------------------

typedef __attribute__((ext_vector_type(2))) float v2f;
typedef __attribute__((ext_vector_type(4))) float v4f;
typedef __attribute__((ext_vector_type(8))) float v8f;

#define C_DIM  10
#define B_DIM  8192
#define F_DIM  256
#define O_DIM  64
#define DK_DIM 5

#define P1_BLOCKS 32
#define P1_CHUNK  (B_DIM / P1_BLOCKS)   // 256 rows of x per block

#define MT 128   // M rows per block in phase 3 (8 waves x 16)
#define KT 128   // K tile staged in LDS (128*64*4B = 32 KB)

// Paired-K LDS addressing: element (k, o) -> ((k>>1)*O_DIM + o)*2 + (k&1).
// Rows 2j and 2j+1 of one column are adjacent -> ds_load_b64 per B-fragment.
__device__ __forceinline__ int lds_pk(int k, int o) {
    return (((k >> 1) * O_DIM + o) << 1) | (k & 1);
}

// ---------------------------------------------------------------------------
// Phase 1: partial[blk][c][f] = sum_{b in chunk} y[b,c] * x[b,f]
// Fixed-order, no atomics -> deterministic.
// ---------------------------------------------------------------------------
__global__ void __launch_bounds__(F_DIM)
phase1_colsum(const float* __restrict__ x,
              const float* __restrict__ y,
              float* __restrict__ partial) {
    const int f  = threadIdx.x;                 // one feature per thread
    const int b0 = blockIdx.x * P1_CHUNK;

    float acc[C_DIM];
#pragma unroll
    for (int c = 0; c < C_DIM; ++c) acc[c] = 0.0f;

    for (int b = 0; b < P1_CHUNK; ++b) {
        const float xv = x[(size_t)(b0 + b) * F_DIM + f];
        const float* yb = y + (size_t)(b0 + b) * C_DIM;   // uniform -> scalar loads
#pragma unroll
        for (int c = 0; c < C_DIM; ++c) acc[c] = fmaf(yb[c], xv, acc[c]);
    }

#pragma unroll
    for (int c = 0; c < C_DIM; ++c)
        partial[((size_t)blockIdx.x * C_DIM + c) * F_DIM + f] = acc[c];
}

// ---------------------------------------------------------------------------
// Phase 2: score[c,f] = (s_c/B) * colsum[c,f];  s_c = dot(w_key[c], w_query[c])
// attn/maxnorm algebra: softmax(s)/max(softmax(s)) == exp(s - max(s)).
// Store attn_s[c,f] = exp(score - max_f score) * w_value[c]  (value scale folded).
// ---------------------------------------------------------------------------
__global__ void __launch_bounds__(F_DIM)
phase2_attn(const float* __restrict__ partial,
            const float* __restrict__ w_key,
            const float* __restrict__ w_query,
            const float* __restrict__ w_value,
            float* __restrict__ attn_s) {
    __shared__ float red[F_DIM];
    const int f = threadIdx.x;

    float score[C_DIM];
#pragma unroll
    for (int c = 0; c < C_DIM; ++c) {
        float s = 0.0f;
        for (int blk = 0; blk < P1_BLOCKS; ++blk)
            s += partial[((size_t)blk * C_DIM + c) * F_DIM + f];
        float sc = 0.0f;
#pragma unroll
        for (int k = 0; k < DK_DIM; ++k)
            sc = fmaf(w_key[c * DK_DIM + k], w_query[c * DK_DIM + k], sc);
        score[c] = s * (sc * (1.0f / (float)B_DIM));
    }

    for (int c = 0; c < C_DIM; ++c) {
        red[f] = score[c];
        __syncthreads();
        for (int off = F_DIM / 2; off > 0; off >>= 1) {
            if (f < off) red[f] = fmaxf(red[f], red[f + off]);
            __syncthreads();
        }
        const float m = red[0];
        __syncthreads();
        attn_s[c * F_DIM + f] = expf(score[c] - m) * w_value[c];
    }
}

// ---------------------------------------------------------------------------
// Phase 3: out[c,b,o] = sum_f x[b,f] * (attn_s[c,f]*weight[c,f,o]) + bias[c,o]
// Per block: c = blockIdx.z, 128 rows of x. 8 waves, each wave computes a
// 16(M) x 64(N) strip via chained v_wmma_f32_16x16x4_f32.
// A-frag layout (32-bit A 16x4): lane<16 -> M=lane, K={0,1}; lane>=16 -> K={2,3}.
// B-frag layout (32-bit B 4x16, mirror): v0 = B[K0][n], v1 = B[K0+1][n].
// D layout: VGPR j -> M = j + (lane>=16 ? 8 : 0), N = lane&15.
// ---------------------------------------------------------------------------
__global__ void __launch_bounds__(256)
phase3_gemm(const float* __restrict__ x,
            const float* __restrict__ weight,
            const float* __restrict__ bias,
            const float* __restrict__ attn_s,
            float* __restrict__ out) {
    __shared__ float lW[KT * O_DIM];            // 32 KB scaled-weight tile, paired-K

    const int c     = blockIdx.z;
    const int tid   = threadIdx.x;
    const int lane  = tid & 31;
    const int wave  = tid >> 5;                 // 8 waves per block
    const int laneM = lane & 15;
    const int laneK = (lane >> 4) * 2;          // 0 or 2
    const int m0    = blockIdx.x * MT + wave * 16;

    const float* wc   = weight + (size_t)c * F_DIM * O_DIM;
    const float* ac   = attn_s + (size_t)c * F_DIM;
    const float* xrow = x + (size_t)(m0 + laneM) * F_DIM;
    __builtin_prefetch(xrow + 128, 0, 1);       // global_prefetch_b8

    const v8f zero = {0.f, 0.f, 0.f, 0.f, 0.f, 0.f, 0.f, 0.f};
    v8f acc0 = zero, acc1 = zero, acc2 = zero, acc3 = zero;

    for (int kt = 0; kt < F_DIM; kt += KT) {
        __syncthreads();
        // Stage attn-scaled weight tile into paired-K LDS layout.
        // KT*O_DIM = 8192 floats; 8x v4f reads per thread, scatter stride-2 writes.
#pragma unroll
        for (int r = 0; r < (KT * O_DIM) / (256 * 4); ++r) {
            const int e  = (r * 256 + tid) * 4;
            const int fl = e >> 6;              // local k row (4 o's share one f)
            const int o0 = e & 63;
            const float s = ac[kt + fl];
            v4f w4 = *reinterpret_cast<const v4f*>(wc + (size_t)(kt + fl) * O_DIM + o0);
            w4 *= s;
#pragma unroll
            for (int i = 0; i < 4; ++i)
                lW[lds_pk(fl, o0 + i)] = w4[i];
        }
        __syncthreads();

#pragma unroll 8
        for (int k = 0; k < KT; k += 4) {
            // A fragment: two consecutive K values for this lane's row.
            const v2f a  = *reinterpret_cast<const v2f*>(xrow + kt + k + laneK);
            // B fragments: contiguous 8B pairs in paired-K layout -> ds_load_b64.
            const int kb = lds_pk(k + laneK, laneM);       // even index, 8B aligned
            const v2f b0 = *reinterpret_cast<const v2f*>(&lW[kb +  0 * 32]);
            const v2f b1 = *reinterpret_cast<const v2f*>(&lW[kb +  1 * 32]);
            const v2f b2 = *reinterpret_cast<const v2f*>(&lW[kb +  2 * 32]);
            const v2f b3 = *reinterpret_cast<const v2f*>(&lW[kb +  3 * 32]);
            acc0 = __builtin_amdgcn_wmma_f32_16x16x4_f32(false, a, false, b0, (short)0, acc0, false, false);
            acc1 = __builtin_amdgcn_wmma_f32_16x16x4_f32(false, a, false, b1, (short)0, acc1, false, false);
            acc2 = __builtin_amdgcn_wmma_f32_16x16x4_f32(false, a, false, b2, (short)0, acc2, false, false);
            acc3 = __builtin_amdgcn_wmma_f32_16x16x4_f32(false, a, false, b3, (short)0, acc3, false, false);
        }
    }

    // Bias add + store per documented 16x16 f32 D layout.
    const int rbase = m0 + (lane >> 4) * 8;
    const v8f accs[4] = {acc0, acc1, acc2, acc3};
#pragma unroll
    for (int t = 0; t < 4; ++t) {
        const int col  = t * 16 + laneM;
        const float bv = bias[c * O_DIM + col];
#pragma unroll
        for (int j = 0; j < 8; ++j) {
            out[((size_t)c * B_DIM + (size_t)(rbase + j)) * O_DIM + col] = accs[t][j] + bv;
        }
    }
}

// ---------------------------------------------------------------------------
// Launch: inputs in setup_inputs() order: x, y, w_key, w_query, w_value,
// weight, bias. Output: [C,B,O] f32. Workspace: partials + attn_s (~338 KB).
// ---------------------------------------------------------------------------
extern "C" void kernel_launch(void* const* d_in, const int* in_sizes, int n_in,
                              void* d_out, int out_size, void* d_ws, size_t ws_size,
                              hipStream_t stream) {
    (void)in_sizes; (void)n_in; (void)out_size; (void)ws_size;
    const float* x   = (const float*)d_in[0];
    const float* y   = (const float*)d_in[1];
    const float* wk  = (const float*)d_in[2];
    const float* wq  = (const float*)d_in[3];
    const float* wv  = (const float*)d_in[4];
    const float* wgt = (const float*)d_in[5];
    const float* bs  = (const float*)d_in[6];
    float* out = (float*)d_out;

    float* ws      = (float*)d_ws;
    float* partial = ws;                                     // 32*10*256 floats
    float* attn_s  = ws + (size_t)P1_BLOCKS * C_DIM * F_DIM; // 10*256 floats

    phase1_colsum<<<dim3(P1_BLOCKS), dim3(F_DIM), 0, stream>>>(x, y, partial);
    phase2_attn  <<<dim3(1),         dim3(F_DIM), 0, stream>>>(partial, wk, wq, wv, attn_s);
    phase3_gemm  <<<dim3(B_DIM / MT, 1, C_DIM), dim3(256), 0, stream>>>(x, wgt, bs, attn_s, out);
}